// GATBlock_33904471835030
// MI455X (gfx1250) — compile-verified
//
#include <hip/hip_runtime.h>
#include <hip/hip_bf16.h>
#include <math.h>

#define N_NODESC 50000
#define N_EDGESC 800000
#define HF       64
#define N_GRAPHS 512
#define NEG_SLOPE 0.2f
#define GN_EPS    1e-5f
#define N_TOT    (N_EDGESC + N_NODESC)

typedef __attribute__((ext_vector_type(2))) float v2f;
typedef __attribute__((ext_vector_type(8))) float v8f;

// Order-preserving IEEE-754 -> u32 encoding so atomicMax(u32) == float max.
static __device__ __forceinline__ unsigned encf(float x) {
  unsigned u = __float_as_uint(x);
  return (u & 0x80000000u) ? ~u : (u | 0x80000000u);
}
static __device__ __forceinline__ float decf(unsigned u) {
  return (u & 0x80000000u) ? __uint_as_float(u & 0x7fffffffu) : __uint_as_float(~u);
}
#define ENC_NEG_INF 0x007FFFFFu   // encf(-inf)

__global__ void k_fill(unsigned* __restrict__ p, int n, unsigned v) {
  int t = blockIdx.x * blockDim.x + threadIdx.x;
  if (t < n) p[t] = v;
}

// Per edge: degree + edge_attr scatter-sum by dst (for self-loop fill 'mean').
__global__ void k_deg_ea(const int* __restrict__ dst, const float* __restrict__ ea,
                         float* __restrict__ deg, float* __restrict__ ea_sum) {
  int e = blockIdx.x * blockDim.x + threadIdx.x;
  if (e >= N_EDGESC) return;
  int d = dst[e];
  atomicAdd(deg + d, 1.0f);
  atomicAdd(ea_sum + d * 3 + 0, ea[e * 3 + 0]);
  atomicAdd(ea_sum + d * 3 + 1, ea[e * 3 + 1]);
  atomicAdd(ea_sum + d * 3 + 2, ea[e * 3 + 2]);
}

__global__ void k_loop_attr(float* __restrict__ ea_sum, const float* __restrict__ deg) {
  int n = blockIdx.x * blockDim.x + threadIdx.x;
  if (n >= N_NODESC) return;
  float inv = 1.0f / fmaxf(deg[n], 1.0f);
  ea_sum[n * 3 + 0] *= inv;
  ea_sum[n * 3 + 1] *= inv;
  ea_sum[n * 3 + 2] *= inv;
}

// xl = x@Wl + bl, xr = x@Wr + br via V_WMMA_F32_16X16X4_F32 (fp32 in/accum).
// One wave per 16-row tile; A operand (x tile) reused for both weight matrices.
// f32 16x4 A layout: VGPR0 = K {0 | 2}, VGPR1 = K {1 | 3} (lanes 0-15 | 16-31).
__global__ void __launch_bounds__(128)
k_gemm_xlr(const float* __restrict__ x,
           const float* __restrict__ Wl, const float* __restrict__ bl,
           const float* __restrict__ Wr, const float* __restrict__ br,
           float* __restrict__ xl, float* __restrict__ xr) {
  const int lane = threadIdx.x & 31;
  const int wave = threadIdx.x >> 5;
  const int tile = blockIdx.x * 4 + wave;          // wave-uniform -> EXEC all-ones
  if (tile >= N_NODESC / 16) return;
  const int m0 = tile * 16;
  const int lr = lane & 15;
  const int kh = (lane >> 4) << 1;                 // 0 or 2
  const float* __restrict__ xrow = x + (size_t)(m0 + lr) * HF;
#pragma unroll
  for (int nt = 0; nt < 4; ++nt) {
    const int ncol = nt * 16 + lr;
    v8f cl = {0.f, 0.f, 0.f, 0.f, 0.f, 0.f, 0.f, 0.f};
    v8f cr = {0.f, 0.f, 0.f, 0.f, 0.f, 0.f, 0.f, 0.f};
#pragma unroll
    for (int k = 0; k < HF; k += 4) {
      v2f a;  a.x  = xrow[k + kh];              a.y  = xrow[k + kh + 1];
      v2f b0; b0.x = Wl[(k + kh) * HF + ncol];  b0.y = Wl[(k + kh + 1) * HF + ncol];
      v2f b1; b1.x = Wr[(k + kh) * HF + ncol];  b1.y = Wr[(k + kh + 1) * HF + ncol];
      cl = __builtin_amdgcn_wmma_f32_16x16x4_f32(false, a, false, b0, (short)0, cl, false, false);
      cr = __builtin_amdgcn_wmma_f32_16x16x4_f32(false, a, false, b1, (short)0, cr, false, false);
    }
    const int mb = m0 + ((lane >> 4) << 3);      // C/D: lanes 16-31 hold M=8..15
    const float bLn = bl[ncol], bRn = br[ncol];
#pragma unroll
    for (int r = 0; r < 8; ++r) {
      xl[(size_t)(mb + r) * HF + ncol] = cl[r] + bLn;
      xr[(size_t)(mb + r) * HF + ncol] = cr[r] + bRn;
    }
  }
}

// score_i = att . leaky_relu(xl[src] + xr[dst] + ea@We); segment-max by dst.
__global__ void k_score_max(const float* __restrict__ xl, const float* __restrict__ xr,
                            const float* __restrict__ We, const float* __restrict__ att,
                            const float* __restrict__ eattr, const float* __restrict__ loop_attr,
                            const int* __restrict__ srcA, const int* __restrict__ dstA,
                            float* __restrict__ score, unsigned* __restrict__ smax) {
  int i = blockIdx.x * blockDim.x + threadIdx.x;
  if (i >= N_TOT) return;
  int s, d; const float* a3;
  if (i < N_EDGESC) { s = srcA[i]; d = dstA[i]; a3 = eattr + (size_t)i * 3; }
  else              { s = d = i - N_EDGESC;     a3 = loop_attr + (size_t)(i - N_EDGESC) * 3; }
  const float e0 = a3[0], e1 = a3[1], e2 = a3[2];
  const float* __restrict__ pl = xl + (size_t)s * HF;
  const float* __restrict__ pr = xr + (size_t)d * HF;
  float acc = 0.f;
#pragma unroll 8
  for (int h = 0; h < HF; ++h) {
    float m = pl[h] + pr[h] + e0 * We[h] + e1 * We[HF + h] + e2 * We[2 * HF + h];
    float l = (m > 0.f) ? m : NEG_SLOPE * m;
    acc = fmaf(att[h], l, acc);
  }
  score[i] = acc;
  atomicMax(smax + d, encf(acc));
}

__global__ void k_z_denom(const int* __restrict__ dstA, float* __restrict__ z,
                          const unsigned* __restrict__ smax, float* __restrict__ denom) {
  int i = blockIdx.x * blockDim.x + threadIdx.x;
  if (i >= N_TOT) return;
  int d = (i < N_EDGESC) ? dstA[i] : (i - N_EDGESC);
  float zz = expf(z[i] - decf(smax[d]));
  z[i] = zz;
  atomicAdd(denom + d, zz);
}

__global__ void k_invden(const float* __restrict__ denom, float* __restrict__ invd) {
  int n = blockIdx.x * blockDim.x + threadIdx.x;
  if (n < N_NODESC) invd[n] = 1.0f / denom[n];   // self-loop guarantees denom > 0
}

// out[dst] += alpha * xl[src]; 4 feats/thread, float4 gather + 4 f32 atomics.
__global__ void k_aggregate(const int* __restrict__ srcA, const int* __restrict__ dstA,
                            const float* __restrict__ z, const float* __restrict__ invd,
                            const float* __restrict__ xl, float* __restrict__ out) {
  int t = blockIdx.x * blockDim.x + threadIdx.x;
  if (t >= N_TOT * 16) return;
  int i = t >> 4, h4 = (t & 15) << 2;
  int s, d;
  if (i < N_EDGESC) { s = srcA[i]; d = dstA[i]; } else { s = d = i - N_EDGESC; }
  float alpha = z[i] * invd[d];
  const float4 xv = *reinterpret_cast<const float4*>(xl + (size_t)s * HF + h4);
  float* po = out + (size_t)d * HF + h4;
  atomicAdd(po + 0, alpha * xv.x);
  atomicAdd(po + 1, alpha * xv.y);
  atomicAdd(po + 2, alpha * xv.z);
  atomicAdd(po + 3, alpha * xv.w);
}

__global__ void k_bias_mean(float* __restrict__ out, const float* __restrict__ bias,
                            const int* __restrict__ batch, float* __restrict__ meansum,
                            float* __restrict__ cnt) {
  int t = blockIdx.x * blockDim.x + threadIdx.x;
  if (t >= N_NODESC * HF) return;
  int n = t >> 6, h = t & 63;
  int g = batch[n];
  float v = out[t] + bias[h];
  out[t] = v;
  atomicAdd(meansum + g * HF + h, v);
  if (h == 0) atomicAdd(cnt + g, 1.0f);
}

__global__ void k_mean_fin(float* __restrict__ meansum, const float* __restrict__ cnt) {
  int t = blockIdx.x * blockDim.x + threadIdx.x;
  if (t < N_GRAPHS * HF) meansum[t] /= fmaxf(cnt[t >> 6], 1.0f);
}

__global__ void k_center_var(float* __restrict__ out, const float* __restrict__ mean,
                             const float* __restrict__ ga, const int* __restrict__ batch,
                             float* __restrict__ varsum) {
  int t = blockIdx.x * blockDim.x + threadIdx.x;
  if (t >= N_NODESC * HF) return;
  int n = t >> 6, h = t & 63;
  int g = batch[n];
  float c = out[t] - ga[h] * mean[g * HF + h];
  out[t] = c;
  atomicAdd(varsum + g * HF + h, c * c);
}

__global__ void k_rstd(float* __restrict__ varsum, const float* __restrict__ cnt) {
  int t = blockIdx.x * blockDim.x + threadIdx.x;
  if (t >= N_GRAPHS * HF) return;
  float v = varsum[t] / fmaxf(cnt[t >> 6], 1.0f);
  varsum[t] = rsqrtf(v + GN_EPS);
}

// normalize + ReLU + global max pool (post-ReLU values >= 0, so raw-bit u32 max works).
__global__ void k_final(const float* __restrict__ centered, const float* __restrict__ rstd,
                        const float* __restrict__ gw, const float* __restrict__ gb,
                        const int* __restrict__ batch, float* __restrict__ hemb,
                        unsigned* __restrict__ flat_bits) {
  int t = blockIdx.x * blockDim.x + threadIdx.x;
  if (t >= N_NODESC * HF) return;
  int n = t >> 6, h = t & 63;
  int g = batch[n];
  float hv = gw[h] * centered[t] * rstd[g * HF + h] + gb[h];
  float he = fmaxf(hv, 0.0f);
  hemb[t] = he;
  atomicMax(flat_bits + g * HF + h, __float_as_uint(he));
}

static inline int cdiv(long long a, int b) { return (int)((a + b - 1) / b); }

extern "C" void kernel_launch(void* const* d_in, const int* in_sizes, int n_in,
                              void* d_out, int out_size, void* d_ws, size_t ws_size,
                              hipStream_t stream) {
  const float* x    = (const float*)d_in[0];
  const float* eat  = (const float*)d_in[1];
  const float* Wl   = (const float*)d_in[2];
  const float* bl   = (const float*)d_in[3];
  const float* Wr   = (const float*)d_in[4];
  const float* br   = (const float*)d_in[5];
  const float* We   = (const float*)d_in[6];
  const float* att  = (const float*)d_in[7];
  const float* bias = (const float*)d_in[8];
  const float* gw   = (const float*)d_in[9];
  const float* gb   = (const float*)d_in[10];
  const float* ga   = (const float*)d_in[11];
  const int*   ei   = (const int*)d_in[12];
  const int*   bat  = (const int*)d_in[13];
  const int* srcA = ei;
  const int* dstA = ei + N_EDGESC;

  // ---- workspace layout (contiguous zero-block first) ----
  float* ws      = (float*)d_ws;
  float* out     = ws;                                   // N*64
  float* ea_sum  = out + (size_t)N_NODESC * HF;          // N*3
  float* deg     = ea_sum + (size_t)N_NODESC * 3;        // N
  float* denom   = deg + N_NODESC;                       // N
  float* meansum = denom + N_NODESC;                     // G*64
  float* cnt     = meansum + N_GRAPHS * HF;              // G
  float* varsum  = cnt + N_GRAPHS;                       // G*64 (-> rstd)
  unsigned* smax = (unsigned*)(varsum + N_GRAPHS * HF);  // N
  float* xl      = (float*)(smax + N_NODESC);            // N*64
  float* xr      = xl + (size_t)N_NODESC * HF;           // N*64
  float* z       = xr + (size_t)N_NODESC * HF;           // E+N
  float* invd    = z + N_TOT;                            // N

  // ---- output layout: h_emb | flat | edge_index | edge_attr | batch ----
  float* hemb = (float*)d_out;
  float* flat = hemb + (size_t)N_NODESC * HF;
  int*   o_ei = (int*)(flat + N_GRAPHS * HF);
  float* o_ea = (float*)(o_ei + (size_t)2 * N_EDGESC);
  int*   o_ba = (int*)(o_ea + (size_t)N_EDGESC * 3);

  const int B = 256;
  const long long zero_cnt = (long long)N_NODESC * HF + (long long)N_NODESC * 3
                           + 2LL * N_NODESC + 2LL * N_GRAPHS * HF + N_GRAPHS;
  k_fill<<<cdiv(zero_cnt, B), B, 0, stream>>>((unsigned*)ws, (int)zero_cnt, 0u);
  k_fill<<<cdiv(N_NODESC, B), B, 0, stream>>>(smax, N_NODESC, ENC_NEG_INF);
  k_fill<<<cdiv(N_GRAPHS * HF, B), B, 0, stream>>>((unsigned*)flat, N_GRAPHS * HF, 0u);

  k_deg_ea   <<<cdiv(N_EDGESC, B), B, 0, stream>>>(dstA, eat, deg, ea_sum);
  k_loop_attr<<<cdiv(N_NODESC, B), B, 0, stream>>>(ea_sum, deg);
  k_gemm_xlr <<<cdiv(N_NODESC / 16, 4), 128, 0, stream>>>(x, Wl, bl, Wr, br, xl, xr);
  k_score_max<<<cdiv(N_TOT, B), B, 0, stream>>>(xl, xr, We, att, eat, ea_sum, srcA, dstA, z, smax);
  k_z_denom  <<<cdiv(N_TOT, B), B, 0, stream>>>(dstA, z, smax, denom);
  k_invden   <<<cdiv(N_NODESC, B), B, 0, stream>>>(denom, invd);
  k_aggregate<<<cdiv((long long)N_TOT * 16, B), B, 0, stream>>>(srcA, dstA, z, invd, xl, out);
  k_bias_mean<<<cdiv((long long)N_NODESC * HF, B), B, 0, stream>>>(out, bias, bat, meansum, cnt);
  k_mean_fin <<<cdiv(N_GRAPHS * HF, B), B, 0, stream>>>(meansum, cnt);
  k_center_var<<<cdiv((long long)N_NODESC * HF, B), B, 0, stream>>>(out, meansum, ga, bat, varsum);
  k_rstd     <<<cdiv(N_GRAPHS * HF, B), B, 0, stream>>>(varsum, cnt);
  k_final    <<<cdiv((long long)N_NODESC * HF, B), B, 0, stream>>>(out, varsum, gw, gb, bat,
                                                                   hemb, (unsigned*)flat);

  hipMemcpyAsync(o_ei, ei,  sizeof(int)   * 2 * N_EDGESC, hipMemcpyDeviceToDevice, stream);
  hipMemcpyAsync(o_ea, eat, sizeof(float) * 3 * N_EDGESC, hipMemcpyDeviceToDevice, stream);
  hipMemcpyAsync(o_ba, bat, sizeof(int)   * N_NODESC,     hipMemcpyDeviceToDevice, stream);
}